// BoxQueryAndGroup_74071005986903
// MI455X (gfx1250) — compile-verified
//
#include <hip/hip_runtime.h>
#include <stdint.h>

// ---------------------------------------------------------------------------
// BoxQueryAndGroup for MI455X (gfx1250, wave32).
//
// Reference semantics (replicated exactly):
//   key: idx_f[n] = n if in_box else 0; stable ascending argsort; take 32.
//   => first 32 indices with key 0 (n==0 or !in_box) ascending; if fewer
//      than 32 zeros, in-box indices ascending fill the rest.
//   grouped_xyz = xyz[idx] - center            -> (B,3,NQ,32)
//   new_features = concat(grouped_xyz, F[idx]) -> (B,67,NQ,32)
//   local_group_mask == all False              -> (B,NQ,32) zeros
//
// Problem is latency-bound gather (~26MB total traffic, ~1us at 23.3TB/s).
// No matmul exists -> WMMA not applicable. CDNA5 paths used instead:
//   - wave32 ballot compaction (one wave per (b,q) query)
//   - global_load_async_to_lds_b32 DMA staging of xyz chunks (ASYNCcnt)
//   - split s_wait_dscnt / s_wait_asynccnt counters
// ---------------------------------------------------------------------------

#define B_       4
#define N_       16384
#define C_       64
#define NQ_      256
#define NS_      32

#define GX_ELEMS ((size_t)B_ * 3 * NQ_ * NS_)          // 98304
#define NF_ELEMS ((size_t)B_ * (3 + C_) * NQ_ * NS_)   // 2195456
#define MK_ELEMS ((size_t)B_ * NQ_ * NS_)              // 32768

// Stage 32 points (96 dwords, contiguous 384B) of xyz into LDS via the
// async DMA path. Must be called with all 32 lanes active (uniform flow).
// Caller guarantees prior LDS reads of s_xyz are consumed (we wait dscnt
// here before overwriting).
__device__ __forceinline__ void stage_chunk_async(const float* gbase,
                                                  float* s_xyz,
                                                  unsigned lane) {
    // Ensure previous ds_loads of this buffer have completed before the DMA
    // engine overwrites it.
    asm volatile("s_wait_dscnt 0x0" ::: "memory");
#pragma unroll
    for (int k = 0; k < 3; ++k) {
        // Generic LDS pointer low 32 bits == wave-relative LDS byte offset
        // (flat->LDS aperture mapping truncates the upper bits).
        unsigned lds_off = (unsigned)(uintptr_t)(&s_xyz[k * 32 + lane]);
        const float* gp = gbase + k * 32 + lane;
        asm volatile("global_load_async_to_lds_b32 %0, %1, off"
                     :: "v"(lds_off), "v"(gp)
                     : "memory");
    }
    asm volatile("s_wait_asynccnt 0x0" ::: "memory");
}

__global__ __launch_bounds__(32)
void BoxQueryAndGroup_kernel(const float* __restrict__ key_xyz,     // (B,N,3)
                             const float* __restrict__ key_feat,    // (B,C,N)
                             const float* __restrict__ query_box,   // (B,NQ,6)
                             float* __restrict__ out) {
    __shared__ float s_xyz[96];      // 32 points x 3 floats staging buffer
    __shared__ int   s_idx[NS_];     // selected indices for this query

    const int w = blockIdx.x;        // 0 .. B*NQ-1, one wave per query
    const int b = w / NQ_;
    const int q = w - b * NQ_;
    const unsigned lane = threadIdx.x;      // 0..31 (wave32)
    const unsigned lt_mask = (1u << lane) - 1u;

    const float* qb = query_box + ((size_t)b * NQ_ + q) * 6;
    const float cx = qb[0], cy = qb[1], cz = qb[2];
    const float hx = qb[3] * 0.5f, hy = qb[4] * 0.5f, hz = qb[5] * 0.5f;

    const float* xyz_b = key_xyz + (size_t)b * N_ * 3;

    // ---- phase 1: indices with sort-key 0 (n==0 or not-in-box), ascending
    int count = 0;   // zeros seen so far (uniform across the wave)
    for (int base = 0; base < N_ && count < NS_; base += 32) {
        stage_chunk_async(xyz_b + (size_t)base * 3, s_xyz, lane);
        const float px = s_xyz[lane * 3 + 0];
        const float py = s_xyz[lane * 3 + 1];
        const float pz = s_xyz[lane * 3 + 2];
        const int n = base + (int)lane;
        const bool inb = (__builtin_fabsf(px - cx) <= hx) &&
                         (__builtin_fabsf(py - cy) <= hy) &&
                         (__builtin_fabsf(pz - cz) <= hz);
        const bool zero = (n == 0) || (!inb);
        const unsigned m = __builtin_amdgcn_ballot_w32(zero);
        if (zero) {
            const int pos = count + (int)__builtin_popcount(m & lt_mask);
            if (pos < NS_) s_idx[pos] = n;
        }
        count += (int)__builtin_popcount(m);
    }

    // ---- phase 2 (rare): fewer than 32 zero-key points; append in-box
    //      indices ascending (key value == index for these).
    if (count < NS_) {
        int cnt2 = count;
        for (int base = 0; base < N_ && cnt2 < NS_; base += 32) {
            stage_chunk_async(xyz_b + (size_t)base * 3, s_xyz, lane);
            const float px = s_xyz[lane * 3 + 0];
            const float py = s_xyz[lane * 3 + 1];
            const float pz = s_xyz[lane * 3 + 2];
            const int n = base + (int)lane;
            const bool inb = (n > 0) &&
                             (__builtin_fabsf(px - cx) <= hx) &&
                             (__builtin_fabsf(py - cy) <= hy) &&
                             (__builtin_fabsf(pz - cz) <= hz);
            const unsigned m = __builtin_amdgcn_ballot_w32(inb);
            if (inb) {
                const int pos = cnt2 + (int)__builtin_popcount(m & lt_mask);
                if (pos < NS_) s_idx[pos] = n;
            }
            cnt2 += (int)__builtin_popcount(m);
        }
    }

    // Hand off s_idx (DS stores by scattered lanes -> reads by all lanes).
    asm volatile("s_wait_dscnt 0x0" ::: "memory");
    __builtin_amdgcn_wave_barrier();

    const int i = s_idx[lane];       // this lane's sample index

    // ---- gather + write outputs (lane == sample slot s, stores coalesced)
    const float px = xyz_b[(size_t)i * 3 + 0];
    const float py = xyz_b[(size_t)i * 3 + 1];
    const float pz = xyz_b[(size_t)i * 3 + 2];
    const float gx = px - cx, gy = py - cy, gz = pz - cz;

    float* out_gx = out;
    float* out_nf = out + GX_ELEMS;
    float* out_mk = out + GX_ELEMS + NF_ELEMS;

    const size_t plane = (size_t)NQ_ * NS_;          // stride between channels
    const size_t qs = (size_t)q * NS_ + lane;

    // grouped_xyz (B,3,NQ,NS)
    out_gx[((size_t)b * 3 + 0) * plane + qs] = gx;
    out_gx[((size_t)b * 3 + 1) * plane + qs] = gy;
    out_gx[((size_t)b * 3 + 2) * plane + qs] = gz;

    // new_features (B,67,NQ,NS): channels 0..2 = grouped_xyz
    const size_t nf0 = (size_t)b * (3 + C_) * plane + qs;
    out_nf[nf0 + 0 * plane] = gx;
    out_nf[nf0 + 1 * plane] = gy;
    out_nf[nf0 + 2 * plane] = gz;

    // channels 3..66 = key_features[b, c, i]  (L2-resident gather)
    const float* fb = key_feat + (size_t)b * C_ * N_ + (size_t)i;
#pragma unroll 8
    for (int c = 0; c < C_; ++c) {
        out_nf[nf0 + (size_t)(3 + c) * plane] = fb[(size_t)c * N_];
    }

    // local_group_mask: provably all False (index 0 sorts to slot 0, which
    // the reference forces to False; a permutation has no other zeros).
    out_mk[((size_t)b * NQ_ + q) * NS_ + lane] = 0.0f;
}

extern "C" void kernel_launch(void* const* d_in, const int* in_sizes, int n_in,
                              void* d_out, int out_size, void* d_ws, size_t ws_size,
                              hipStream_t stream) {
    (void)in_sizes; (void)n_in; (void)out_size; (void)d_ws; (void)ws_size;
    const float* key_xyz   = (const float*)d_in[0];   // (4,16384,3) f32
    const float* key_feat  = (const float*)d_in[1];   // (4,64,16384) f32
    const float* query_box = (const float*)d_in[2];   // (4,256,6) f32
    float* out = (float*)d_out;                       // 2326528 f32

    dim3 grid(B_ * NQ_);   // 1024 single-wave workgroups (one per query)
    dim3 block(32);        // wave32
    BoxQueryAndGroup_kernel<<<grid, block, 0, stream>>>(key_xyz, key_feat,
                                                        query_box, out);
}